// VectorQuantizer_48387101557426
// MI455X (gfx1250) — compile-verified
//
#include <hip/hip_runtime.h>

// VQ-VAE vector quantizer forward for MI455X (gfx1250, wave32).
//
// distances = ||z||^2 + ||c||^2 - 2 z.c ; argmin over codes.
// ||z||^2 is constant per row -> dropped. ||c||^2 folded into the WMMA
// C operand, A pre-scaled by -2, so each v_wmma_f32_16x16x4_f32 chain
// yields argmin-comparable distances directly in fp32 (matches the f32
// reference; lower-precision WMMA would flip argmin on ~1e-5 gaps).

typedef __attribute__((ext_vector_type(2))) float v2f;
typedef __attribute__((ext_vector_type(4))) float v4f;
typedef __attribute__((ext_vector_type(8))) float v8f;

#define K_CODES   1024
#define DIM       64
#define HW        4096      // H*W
#define BSTRIDE   262144    // D*H*W (per-batch stride in z)
#define ZQ_NUMEL  8388608   // 32*64*64*64
#define LOSS_OFF  8388608
#define IDX_OFF   8388609
#define LOSS_SCALE (1.25f / 8388608.0f)  // (1 + 0.25) * mean

// ---- prep: codebook squared norms into workspace; zero loss slot ----
__global__ __launch_bounds__(256) void vq_prep(const float* __restrict__ cb,
                                               float* __restrict__ cnorm,
                                               float* __restrict__ out) {
  int k = blockIdx.x * 256 + threadIdx.x;
  if (k == 0) out[LOSS_OFF] = 0.0f;
  if (k < K_CODES) {
    const v4f* row = (const v4f*)(cb + (size_t)k * DIM);
    float s = 0.0f;
#pragma unroll
    for (int i = 0; i < DIM / 4; ++i) {
      v4f v = row[i];
      s += v.x * v.x + v.y * v.y + v.z * v.z + v.w * v.w;
    }
    cnorm[k] = s;
  }
}

// ---- main: 1 wave handles 16 rows vs all 1024 codes ----
__global__ __launch_bounds__(256) void vq_main(const float* __restrict__ z,
                                               const float* __restrict__ cb,
                                               const float* __restrict__ cnorm,
                                               float* __restrict__ out) {
  __shared__ int s_idx[8][16];

  const int lane   = threadIdx.x & 31;
  const int w      = threadIdx.x >> 5;           // wave in block (8 waves)
  const int gw     = blockIdx.x * 8 + w;         // global wave id
  const int rowBase = gw * 16;                   // 16 z_flat rows per wave
  const int bb     = rowBase / HW;               // batch index (tile never crosses)
  const int pBase  = rowBase % HW;               // h*W + w position
  const int m      = lane & 15;                  // M (A) / N (B,C,D) index
  const int h      = lane >> 4;                  // half: K-pair select

  // A operand (16x4 f32 layout): lane = row M, VGPR pair = K={2h, 2h+1}.
  // Preload all 16 K-steps of -2*z for this 16x64 tile (32 VGPRs).
  const float* zb = z + (size_t)bb * BSTRIDE + pBase + m;
  v2f a[16];
#pragma unroll
  for (int s = 0; s < 16; ++s) {
    int d0 = 4 * s + 2 * h;
    a[s].x = -2.0f * zb[(size_t)d0 * HW];
    a[s].y = -2.0f * zb[(size_t)(d0 + 1) * HW];
  }

  float best[8];
  int   bidx[8];
#pragma unroll
  for (int r = 0; r < 8; ++r) { best[r] = 1e30f; bidx[r] = 0; }

  for (int j = 0; j < 64; ++j) {                 // 64 code tiles of 16
    const int codeBase = j << 4;
    // C init = ||c||^2 of this lane's code column (same for all 8 rows).
    float cn = cnorm[codeBase + m];
    v8f acc;
#pragma unroll
    for (int r = 0; r < 8; ++r) acc[r] = cn;

    // B operand (4x16 f32 layout): lane = code N, VGPR pair = K={2h,2h+1}.
    const float* cbp = cb + (size_t)(codeBase + m) * DIM + 2 * h;
    if (j < 63) __builtin_prefetch(cbp + 16 * DIM, 0, 0);  // next tile
#pragma unroll
    for (int s = 0; s < 16; ++s) {
      v2f bv = *(const v2f*)(cbp + 4 * s);
      acc = __builtin_amdgcn_wmma_f32_16x16x4_f32(
          false, a[s], false, bv, (short)0, acc, false, false);
    }

    // Running per-lane argmin (tiles scanned in increasing code order:
    // strict '<' keeps the lowest index on ties, matching jnp.argmin).
#pragma unroll
    for (int r = 0; r < 8; ++r) {
      if (acc[r] < best[r]) { best[r] = acc[r]; bidx[r] = codeBase + m; }
    }
  }

  // Butterfly argmin across the 16 lanes of each half (D layout:
  // VGPR r <-> row r + 8*h, lanes within half <-> code columns).
#pragma unroll
  for (int r = 0; r < 8; ++r) {
#pragma unroll
    for (int off = 1; off < 16; off <<= 1) {
      float od = __shfl_xor(best[r], off, 32);
      int   oi = __shfl_xor(bidx[r], off, 32);
      if (od < best[r] || (od == best[r] && oi < bidx[r])) {
        best[r] = od; bidx[r] = oi;
      }
    }
  }

  if (lane == 0) {
#pragma unroll
    for (int r = 0; r < 8; ++r) s_idx[w][r] = bidx[r];        // rows 0..7
  } else if (lane == 16) {
#pragma unroll
    for (int r = 0; r < 8; ++r) s_idx[w][8 + r] = bidx[r];    // rows 8..15
  }
  __syncthreads();

  // encoding_indices (written as float values into the f32 output buffer)
  if (lane < 16) out[IDX_OFF + rowBase + lane] = (float)s_idx[w][lane];

  // Output z_q (forward value of the straight-through output) + SSE loss.
  // Lane -> (row m2, d parity dh): each store covers 2 contiguous 64B runs.
  const int m2 = lane & 15;
  const int dh = lane >> 4;
  const int myIdx = s_idx[w][m2];
  const float* qrow = cb + (size_t)myIdx * DIM;
  const size_t zb2 = (size_t)bb * BSTRIDE + pBase + m2;
  float sse = 0.0f;
#pragma unroll 8
  for (int dd = 0; dd < 32; ++dd) {
    int d = 2 * dd + dh;
    size_t off = zb2 + (size_t)d * HW;
    float q = qrow[d];
    float diff = q - z[off];
    out[off] = q;
    sse += diff * diff;
  }
#pragma unroll
  for (int offm = 1; offm < 32; offm <<= 1) sse += __shfl_xor(sse, offm, 32);
  if (lane == 0) atomicAdd(out + LOSS_OFF, sse * LOSS_SCALE);
}

extern "C" void kernel_launch(void* const* d_in, const int* in_sizes, int n_in,
                              void* d_out, int out_size, void* d_ws, size_t ws_size,
                              hipStream_t stream) {
  const float* z  = (const float*)d_in[0];    // (32, 64, 64, 64) f32
  const float* cb = (const float*)d_in[1];    // (1024, 64) f32
  float* out   = (float*)d_out;               // [z_q | loss | indices]
  float* cnorm = (float*)d_ws;                // 1024 floats of scratch

  vq_prep<<<4, 256, 0, stream>>>(cb, cnorm, out);
  // 131072 rows / 16 per wave / 8 waves per block = 1024 blocks
  vq_main<<<1024, 256, 0, stream>>>(z, cb, cnorm, out);
}